// ASFF_78417512890590
// MI455X (gfx1250) — compile-verified
//
#include <hip/hip_runtime.h>
#include <math.h>

// ---------------------------------------------------------------------------
// ASFF forward for MI455X (gfx1250, wave32).
// - All matrix work (2D-DFT-as-GEMM and block-diagonal channel mixing) uses
//   V_WMMA_F32_16X16X4_F32: exact fp32 math on the matrix pipe. The problem is
//   L2/bandwidth bound (~18 GFLOP vs ~1.5 GB of traffic at 23.3 TB/s), so the
//   fp32 WMMA rate is more than sufficient and precision is preserved.
// - FFT image staging uses GLOBAL_LOAD_ASYNC_TO_LDS_B128 (ASYNCcnt +
//   s_wait_asynccnt): DMA straight into LDS, no VGPR staging.
// - BN2 is fused into the recombine kernel to remove two full-tensor passes.
// ---------------------------------------------------------------------------

typedef float v2f __attribute__((ext_vector_type(2)));
typedef float v8f __attribute__((ext_vector_type(8)));

#define TT   4
#define BBATCH 8
#define CCH  256
#define HH   48
#define IMG  2304                    // 48*48
#define NSPAT (BBATCH*CCH*IMG)       // 4,718,592  (one timestep)
#define NTOT  ((size_t)TT*NSPAT)     // 18,874,368 (full tensor)

static __device__ __forceinline__ v8f wmma4(v2f a, v2f b, v8f c) {
  // D = A(16x4) * B(4x16) + C, fp32, wave32.
  return __builtin_amdgcn_wmma_f32_16x16x4_f32(false, a, false, b, (short)0, c,
                                               false, false);
}

// LIF step: v = (v + x)/2 ; s = (v >= 0.5) ; v *= (1 - s)   (hard reset)
static __device__ __forceinline__ float lif_step(float& v, float x) {
  v = 0.5f * (v + x);
  float s = (v >= 0.5f) ? 1.0f : 0.0f;
  v *= (1.0f - s);
  return s;
}

// ---------------------------------------------------------------------------
// 1) LIF over time on x -> spikes
__global__ __launch_bounds__(256) void lif_kernel(const float* __restrict__ x,
                                                  float* __restrict__ s) {
  size_t idx = (size_t)blockIdx.x * 256 + threadIdx.x;
  if (idx >= (size_t)NSPAT) return;
  float v = 0.0f;
  for (int t = 0; t < TT; ++t) {
    size_t off = (size_t)t * NSPAT + idx;
    s[off] = lif_step(v, x[off]);
  }
}

// In-place LIF on two buffers (real / imag of the FFT output)
__global__ __launch_bounds__(256) void lif2_kernel(float* __restrict__ a,
                                                   float* __restrict__ b) {
  size_t idx = (size_t)blockIdx.x * 256 + threadIdx.x;
  if (idx >= (size_t)NSPAT) return;
  float va = 0.0f, vb = 0.0f;
  for (int t = 0; t < TT; ++t) {
    size_t off = (size_t)t * NSPAT + idx;
    a[off] = lif_step(va, a[off]);
    b[off] = lif_step(vb, b[off]);
  }
}

// ---------------------------------------------------------------------------
// 2) DFT matrices: Fc = cos(2*pi*u*h/48)/sqrt(48), Fs = -sin(...)/sqrt(48)
//    (two applications of the scaled matrix give the fft2 'ortho' 1/48)
__global__ __launch_bounds__(256) void dft_init_kernel(float* __restrict__ Fc,
                                                       float* __restrict__ Fs) {
  int idx = blockIdx.x * 256 + threadIdx.x;
  if (idx >= IMG) return;
  int u = idx / HH, h = idx % HH;
  int ph = (u * h) % HH;
  float ang = 6.2831853071795864f * (float)ph / 48.0f;
  const float inv = 0.14433756729740643f;  // 1/sqrt(48)
  Fc[idx] = cosf(ang) * inv;
  Fs[idx] = -sinf(ang) * inv;
}

// ---------------------------------------------------------------------------
// Async DMA of one 48x48 fp32 image (9216 B) global -> LDS.
// 576 x b128 transfers, 18 per lane; tracked by ASYNCcnt, fenced below.
static __device__ __forceinline__ void stage_image_async(
    const float* __restrict__ gp, float* lds_s, int lane) {
#pragma unroll
  for (int i = 0; i < 18; ++i) {
    int elt = (i * 32 + lane) * 4;                 // float index, 16B aligned
    unsigned laddr = (unsigned)(size_t)(lds_s + elt);
    const float* gaddr = gp + elt;
    asm volatile("global_load_async_to_lds_b128 %0, %1, off"
                 :
                 : "v"(laddr), "v"(gaddr)
                 : "memory");
  }
  asm volatile("s_wait_asynccnt 0x0" ::: "memory");
}

// ---------------------------------------------------------------------------
// 3) 2D DFT as GEMM, one wave per 48x48 image.
//    Stage1: M = S*F  -> Mr = S*Fc, Mi = S*Fs   (S real)
//    Stage2: X = F*M  -> Xr = Fc*Mr - Fs*Mi ; Xi = Fc*Mi + Fs*Mr
__global__ __launch_bounds__(32) void fft_fwd_kernel(const float* __restrict__ src,
                                                     const float* __restrict__ Fc,
                                                     const float* __restrict__ Fs,
                                                     float* __restrict__ outr,
                                                     float* __restrict__ outi) {
  __shared__ float S[IMG];
  __shared__ float Mr[IMG];
  __shared__ float Mi[IMG];
  const size_t img = blockIdx.x;
  const int lane = threadIdx.x;
  stage_image_async(src + img * IMG, S, lane);
  __syncthreads();
  const int l = lane & 15;
  const int koff = (lane >> 4) << 1;   // 0 for lanes 0-15, 2 for lanes 16-31
  const int rbase8 = (lane >> 4) << 3; // +8 rows for upper half lanes

  // ---- stage 1 ----
  for (int t = 0; t < 9; ++t) {
    int m0 = (t / 3) * 16, n0 = (t % 3) * 16;
    v8f ar = {}, ai = {};
    for (int k0 = 0; k0 < HH; k0 += 4) {
      v2f a;  a.x  = S[(m0 + l) * HH + k0 + koff];
              a.y  = S[(m0 + l) * HH + k0 + koff + 1];
      v2f bc; bc.x = Fc[(k0 + koff) * HH + n0 + l];
              bc.y = Fc[(k0 + koff + 1) * HH + n0 + l];
      v2f bs; bs.x = Fs[(k0 + koff) * HH + n0 + l];
              bs.y = Fs[(k0 + koff + 1) * HH + n0 + l];
      ar = wmma4(a, bc, ar);
      ai = wmma4(a, bs, ai);
    }
    int col = n0 + l;
    for (int r = 0; r < 8; ++r) {
      Mr[(m0 + rbase8 + r) * HH + col] = ar[r];
      Mi[(m0 + rbase8 + r) * HH + col] = ai[r];
    }
  }
  __syncthreads();

  // ---- stage 2 ----
  float* orp = outr + img * IMG;
  float* oip = outi + img * IMG;
  for (int t = 0; t < 9; ++t) {
    int u0 = (t / 3) * 16, v0 = (t % 3) * 16;
    v8f xr = {}, xi = {};
    for (int k0 = 0; k0 < HH; k0 += 4) {
      v2f ac; ac.x = Fc[(u0 + l) * HH + k0 + koff];
              ac.y = Fc[(u0 + l) * HH + k0 + koff + 1];
      v2f ad; ad.x = Fs[(u0 + l) * HH + k0 + koff];
              ad.y = Fs[(u0 + l) * HH + k0 + koff + 1];
      v2f br; br.x = Mr[(k0 + koff) * HH + v0 + l];
              br.y = Mr[(k0 + koff + 1) * HH + v0 + l];
      v2f bi; bi.x = Mi[(k0 + koff) * HH + v0 + l];
              bi.y = Mi[(k0 + koff + 1) * HH + v0 + l];
      v2f adn; adn.x = -ad.x; adn.y = -ad.y;   // fp32 WMMA has no A-neg mod
      xr = wmma4(ac, br, xr);
      xr = wmma4(adn, bi, xr);
      xi = wmma4(ac, bi, xi);
      xi = wmma4(ad, br, xi);
    }
    int col = v0 + l;
    for (int r = 0; r < 8; ++r) {
      orp[(u0 + rbase8 + r) * HH + col] = xr[r];
      oip[(u0 + rbase8 + r) * HH + col] = xi[r];
    }
  }
}

// Second FFT pass: real part only, in-place safe (image fully staged in LDS
// and ASYNCcnt-fenced before any writeback).
__global__ __launch_bounds__(32) void fft_real_kernel(const float* __restrict__ src,
                                                      const float* __restrict__ Fc,
                                                      const float* __restrict__ Fs,
                                                      float* __restrict__ outr) {
  __shared__ float S[IMG];
  __shared__ float Mr[IMG];
  __shared__ float Mi[IMG];
  const size_t img = blockIdx.x;
  const int lane = threadIdx.x;
  stage_image_async(src + img * IMG, S, lane);
  __syncthreads();
  const int l = lane & 15;
  const int koff = (lane >> 4) << 1;
  const int rbase8 = (lane >> 4) << 3;

  for (int t = 0; t < 9; ++t) {
    int m0 = (t / 3) * 16, n0 = (t % 3) * 16;
    v8f ar = {}, ai = {};
    for (int k0 = 0; k0 < HH; k0 += 4) {
      v2f a;  a.x  = S[(m0 + l) * HH + k0 + koff];
              a.y  = S[(m0 + l) * HH + k0 + koff + 1];
      v2f bc; bc.x = Fc[(k0 + koff) * HH + n0 + l];
              bc.y = Fc[(k0 + koff + 1) * HH + n0 + l];
      v2f bs; bs.x = Fs[(k0 + koff) * HH + n0 + l];
              bs.y = Fs[(k0 + koff + 1) * HH + n0 + l];
      ar = wmma4(a, bc, ar);
      ai = wmma4(a, bs, ai);
    }
    int col = n0 + l;
    for (int r = 0; r < 8; ++r) {
      Mr[(m0 + rbase8 + r) * HH + col] = ar[r];
      Mi[(m0 + rbase8 + r) * HH + col] = ai[r];
    }
  }
  __syncthreads();

  float* orp = outr + img * IMG;
  for (int t = 0; t < 9; ++t) {
    int u0 = (t / 3) * 16, v0 = (t % 3) * 16;
    v8f xr = {};
    for (int k0 = 0; k0 < HH; k0 += 4) {
      v2f ac; ac.x = Fc[(u0 + l) * HH + k0 + koff];
              ac.y = Fc[(u0 + l) * HH + k0 + koff + 1];
      v2f adn; adn.x = -Fs[(u0 + l) * HH + k0 + koff];
               adn.y = -Fs[(u0 + l) * HH + k0 + koff + 1];
      v2f br; br.x = Mr[(k0 + koff) * HH + v0 + l];
              br.y = Mr[(k0 + koff + 1) * HH + v0 + l];
      v2f bi; bi.x = Mi[(k0 + koff) * HH + v0 + l];
              bi.y = Mi[(k0 + koff + 1) * HH + v0 + l];
      xr = wmma4(ac, br, xr);
      xr = wmma4(adn, bi, xr);
    }
    int col = v0 + l;
    for (int r = 0; r < 8; ++r)
      orp[(u0 + rbase8 + r) * HH + col] = xr[r];
  }
}

// ---------------------------------------------------------------------------
// 4) Block-diagonal channel mix: out[tb,k,o,hw] = sum_i x[tb,k,i,hw]*w1[k,i,o]
//    One wave per (tb, k, hw-tile): computes both 16-wide o-tiles for both the
//    real and imag streams (32 WMMAs), fusing (a-b) and (a+b) on store.
__global__ __launch_bounds__(128) void mix_kernel(const float* __restrict__ xr,
                                                  const float* __restrict__ xi,
                                                  const float* __restrict__ w1,
                                                  float* __restrict__ om,
                                                  float* __restrict__ op) {
  const int wid  = blockIdx.x * 4 + (threadIdx.x >> 5);  // 36864 jobs exactly
  const int lane = threadIdx.x & 31;
  const int l = lane & 15;
  const int koff = (lane >> 4) << 1;
  const int rbase8 = (lane >> 4) << 3;

  const int ht  = wid % 144;          // hw tile
  const int rem = wid / 144;
  const int k   = rem & 7;            // block index
  const int tb  = rem >> 3;           // T*B index
  const int hw0 = ht * 16;

  const size_t base = ((size_t)tb * CCH + (size_t)k * 32) * IMG;
  const float* xrb = xr + base;
  const float* xib = xi + base;
  const float* wb  = w1 + k * 1024;   // w1[k][i][o], 32x32
  float* omb = om + base;
  float* opb = op + base;

  v8f aA0 = {}, aA1 = {}, aB0 = {}, aB1 = {};
  for (int i0 = 0; i0 < 32; i0 += 4) {
    // A = w1^T tiles: A[m][kk] = w1[k][i0+kk][o0+m]
    v2f a0; a0.x = wb[(i0 + koff) * 32 + l];
            a0.y = wb[(i0 + koff + 1) * 32 + l];
    v2f a1; a1.x = wb[(i0 + koff) * 32 + 16 + l];
            a1.y = wb[(i0 + koff + 1) * 32 + 16 + l];
    // B tiles: B[kk][n] = x[i0+kk][hw0+n]
    v2f br; br.x = xrb[(size_t)(i0 + koff) * IMG + hw0 + l];
            br.y = xrb[(size_t)(i0 + koff + 1) * IMG + hw0 + l];
    v2f bi; bi.x = xib[(size_t)(i0 + koff) * IMG + hw0 + l];
            bi.y = xib[(size_t)(i0 + koff + 1) * IMG + hw0 + l];
    aA0 = wmma4(a0, br, aA0);
    aA1 = wmma4(a1, br, aA1);
    aB0 = wmma4(a0, bi, aB0);
    aB1 = wmma4(a1, bi, aB1);
  }
  for (int r = 0; r < 8; ++r) {
    int o = rbase8 + r;
    size_t i0o = (size_t)o * IMG + hw0 + l;
    size_t i1o = (size_t)(o + 16) * IMG + hw0 + l;
    omb[i0o] = aA0[r] - aB0[r];
    opb[i0o] = aA0[r] + aB0[r];
    omb[i1o] = aA1[r] - aB1[r];
    opb[i1o] = aA1[r] + aB1[r];
  }
}

// ---------------------------------------------------------------------------
// 5) Training-mode BN stats (deterministic: one block per channel).
__global__ __launch_bounds__(256) void bn_stats_kernel(const float* __restrict__ x,
                                                       float* __restrict__ mean,
                                                       float* __restrict__ rstd) {
  const int c = blockIdx.x;
  const int tid = threadIdx.x;
  float s = 0.0f, ss = 0.0f;
  for (int tb = 0; tb < TT * BBATCH; ++tb) {
    const float* p = x + ((size_t)tb * CCH + c) * IMG;
    for (int j = tid; j < IMG; j += 256) {
      float v = p[j];
      s += v;
      ss += v * v;
    }
  }
  __shared__ float sh[512];
  sh[tid] = s;
  sh[256 + tid] = ss;
  __syncthreads();
  for (int o = 128; o > 0; o >>= 1) {
    if (tid < o) {
      sh[tid] += sh[tid + o];
      sh[256 + tid] += sh[256 + tid + o];
    }
    __syncthreads();
  }
  if (tid == 0) {
    const float invn = 1.0f / (float)(TT * BBATCH * IMG);
    float m = sh[0] * invn;
    float var = sh[256] * invn - m * m;
    mean[c] = m;
    rstd[c] = rsqrtf(var + 1e-5f);
  }
}

// Fused BN-normalize + LIF on two buffers, in-place.
__global__ __launch_bounds__(256) void bn_lif_kernel(
    float* __restrict__ A, float* __restrict__ B,
    const float* __restrict__ mA, const float* __restrict__ rA,
    const float* __restrict__ mB, const float* __restrict__ rB,
    const float* __restrict__ gA, const float* __restrict__ bA,
    const float* __restrict__ gB, const float* __restrict__ bB) {
  size_t idx = (size_t)blockIdx.x * 256 + threadIdx.x;
  if (idx >= (size_t)NSPAT) return;
  int c = (int)((idx / IMG) & (CCH - 1));
  float ma = mA[c], sa = rA[c] * gA[c], oa = bA[c];
  float mb = mB[c], sb = rB[c] * gB[c], ob = bB[c];
  float va = 0.0f, vb = 0.0f;
  for (int t = 0; t < TT; ++t) {
    size_t off = (size_t)t * NSPAT + idx;
    float xa = (A[off] - ma) * sa + oa;
    float xb = (B[off] - mb) * sb + ob;
    A[off] = lif_step(va, xa);
    B[off] = lif_step(vb, xb);
  }
}

// BN-normalize only, in-place (used for BN3 on d_out).
__global__ __launch_bounds__(256) void bn_apply_kernel(
    float* __restrict__ X, const float* __restrict__ mean,
    const float* __restrict__ rstd, const float* __restrict__ g,
    const float* __restrict__ b) {
  size_t idx = (size_t)blockIdx.x * 256 + threadIdx.x;
  if (idx >= NTOT) return;
  int c = (int)((idx / IMG) & (CCH - 1));
  X[idx] = (X[idx] - mean[c]) * rstd[c] * g[c] + b[c];
}

// ---------------------------------------------------------------------------
// 6) Fused BN2 + recombine + softshrink:
//    o2r = BN(a2-b2), o2i = BN(a2+b2) computed on the fly (saves two full
//    read+write passes), y = softshrink( so*(o2r-o2i) - se*(o2r+o2i) ).
__global__ __launch_bounds__(256) void y_bn_kernel(
    const float* __restrict__ so, const float* __restrict__ se,
    const float* __restrict__ a2m, const float* __restrict__ a2p,
    const float* __restrict__ mA, const float* __restrict__ rA,
    const float* __restrict__ gA, const float* __restrict__ bA,
    const float* __restrict__ mB, const float* __restrict__ rB,
    const float* __restrict__ gB, const float* __restrict__ bB,
    float* __restrict__ y) {
  size_t idx = (size_t)blockIdx.x * 256 + threadIdx.x;
  if (idx >= NTOT) return;
  int c = (int)((idx / IMG) & (CCH - 1));
  float o2r = (a2m[idx] - mA[c]) * rA[c] * gA[c] + bA[c];
  float o2i = (a2p[idx] - mB[c]) * rB[c] * gB[c] + bB[c];
  float v = so[idx] * (o2r - o2i) - se[idx] * (o2r + o2i);
  const float lam = 0.06f;
  v = (v > lam) ? (v - lam) : ((v < -lam) ? (v + lam) : 0.0f);
  y[idx] = v;
}

// ---------------------------------------------------------------------------
extern "C" void kernel_launch(void* const* d_in, const int* in_sizes, int n_in,
                              void* d_out, int out_size, void* d_ws, size_t ws_size,
                              hipStream_t stream) {
  (void)in_sizes; (void)n_in; (void)out_size;
  const float* x   = (const float*)d_in[0];
  const float* w1  = (const float*)d_in[1];
  // d_in[2] = alpha (only affects the surrogate gradient; forward ignores it)
  const float* g11 = (const float*)d_in[3];
  const float* b11 = (const float*)d_in[4];
  const float* g12 = (const float*)d_in[5];
  const float* b12 = (const float*)d_in[6];
  const float* g21 = (const float*)d_in[7];
  const float* b21 = (const float*)d_in[8];
  const float* g22 = (const float*)d_in[9];
  const float* b22 = (const float*)d_in[10];
  const float* g3  = (const float*)d_in[11];
  const float* b3  = (const float*)d_in[12];
  float* out = (float*)d_out;
  float* ws  = (float*)d_ws;

  const size_t N = NTOT;
  const size_t need = (6 * N + IMG * 2 + 1024) * sizeof(float);
  if (ws_size < need) return;  // insufficient scratch: no-op rather than corrupt

  float* buf0 = ws;            // s0 spikes -> later raw (a2-b2)
  float* buf1 = ws + 1 * N;    // xf.real  -> so spikes
  float* buf2 = ws + 2 * N;    // xf.imag  -> se spikes
  float* buf3 = ws + 3 * N;    // a-b      -> o1r spikes
  float* buf4 = ws + 4 * N;    // a+b      -> o1i spikes
  float* buf5 = ws + 5 * N;    // raw (a2+b2)
  float* Fc = ws + 6 * N;
  float* Fs = Fc + IMG;
  float* mA = Fs + IMG;
  float* rA = mA + 256;
  float* mB = rA + 256;
  float* rB = mB + 256;

  const int ELEM_BLKS  = (int)((NTOT + 255) / 256);        // 73728
  const int SPAT_BLKS  = (NSPAT + 255) / 256;              // 18432
  const int NIMG       = TT * BBATCH * CCH;                // 8192
  const int MIX_BLKS   = (TT * BBATCH * 8 * 144) / 4;      // 9216 (4 waves/blk)

  // 1) LIF on x
  lif_kernel<<<SPAT_BLKS, 256, 0, stream>>>(x, buf0);
  // 2) DFT matrices
  dft_init_kernel<<<(IMG + 255) / 256, 256, 0, stream>>>(Fc, Fs);
  // 3) forward FFT2 (WMMA GEMM, async-LDS staging), one wave per image
  fft_fwd_kernel<<<NIMG, 32, 0, stream>>>(buf0, Fc, Fs, buf1, buf2);
  // 4) LIF on real/imag (in place -> so, se)
  lif2_kernel<<<SPAT_BLKS, 256, 0, stream>>>(buf1, buf2);
  // 5) mix #1 -> (a-b), (a+b)
  mix_kernel<<<MIX_BLKS, 128, 0, stream>>>(buf1, buf2, w1, buf3, buf4);
  // 6) BN1 stats + fused BN+LIF -> o1r, o1i
  bn_stats_kernel<<<CCH, 256, 0, stream>>>(buf3, mA, rA);
  bn_stats_kernel<<<CCH, 256, 0, stream>>>(buf4, mB, rB);
  bn_lif_kernel<<<SPAT_BLKS, 256, 0, stream>>>(buf3, buf4, mA, rA, mB, rB,
                                               g11, b11, g12, b12);
  // 7) mix #2 -> raw (a2-b2), (a2+b2)   (buf0 is dead, reuse it)
  mix_kernel<<<MIX_BLKS, 128, 0, stream>>>(buf3, buf4, w1, buf0, buf5);
  // 8) BN2 stats; BN2-apply is fused into the recombine kernel
  bn_stats_kernel<<<CCH, 256, 0, stream>>>(buf0, mA, rA);
  bn_stats_kernel<<<CCH, 256, 0, stream>>>(buf5, mB, rB);
  // 9) fused BN2 + recombine + softshrink -> d_out
  y_bn_kernel<<<ELEM_BLKS, 256, 0, stream>>>(buf1, buf2, buf0, buf5,
                                             mA, rA, g21, b21,
                                             mB, rB, g22, b22, out);
  // 10) FFT2 real part, in place on d_out
  fft_real_kernel<<<NIMG, 32, 0, stream>>>(out, Fc, Fs, out);
  // 11) BN3 in place on d_out
  bn_stats_kernel<<<CCH, 256, 0, stream>>>(out, mA, rA);
  bn_apply_kernel<<<ELEM_BLKS, 256, 0, stream>>>(out, mA, rA, g3, b3);
}